// GCN_scratch_44890998178564
// MI455X (gfx1250) — compile-verified
//
#include <hip/hip_runtime.h>
#include <hip/hip_bf16.h>
#include <math.h>

#define N_NODES  50000
#define M_NODES  50000
#define N_EDGES  800000
#define NFEAT    300
#define NHID     256
#define NCLS     20
#define N_IDX    5000

typedef float v2f __attribute__((ext_vector_type(2)));
typedef float v8f __attribute__((ext_vector_type(8)));

// ---------------------------------------------------------------------------
// GEMM1: C[50000x256] = X[50000x300] @ W[300x256], fp32 WMMA 16x16x4.
// Block = 256 threads (8 waves). Each wave owns a 16x16 tile.
// Grid = (3125, 2): 16 rows per block, 128 cols per block (8 waves * 16).
// ---------------------------------------------------------------------------
__global__ void __launch_bounds__(256)
gcn_gemm1_wmma(const float* __restrict__ X, const float* __restrict__ W,
               float* __restrict__ C) {
    const int lane    = threadIdx.x & 31;
    const int wave    = threadIdx.x >> 5;                  // 0..7
    const int rowBase = blockIdx.x * 16;
    const int colBase = blockIdx.y * 128 + wave * 16;
    const int m       = lane & 15;                         // A row / B col / D col
    const int kOff    = (lane >> 4) << 1;                  // 0 (lanes 0-15) or 2

    v8f acc = {};
    const float* xrow = X + (size_t)(rowBase + m) * NFEAT;

    for (int k = 0; k < NFEAT; k += 4) {
        const int ka = k + kOff;
        v2f a;                                             // A 16x4: K=ka, ka+1
        a.x = xrow[ka];
        a.y = xrow[ka + 1];
        v2f b;                                             // B 4x16: rows ka, ka+1
        b.x = W[(size_t)ka * NHID + colBase + m];
        b.y = W[(size_t)(ka + 1) * NHID + colBase + m];
        acc = __builtin_amdgcn_wmma_f32_16x16x4_f32(
            /*neg_a=*/false, a, /*neg_b=*/false, b,
            /*c_mod=*/(short)0, acc, /*reuse_a=*/false, /*reuse_b=*/false);
    }

    // D layout: VGPR r -> M = r + 8*(lane>=16), N = lane&15
    const int mOut = (lane >> 4) * 8;
    #pragma unroll
    for (int r = 0; r < 8; ++r) {
        C[(size_t)(rowBase + mOut + r) * NHID + colBase + m] = acc[r];
    }
}

// ---------------------------------------------------------------------------
// GEMM2: C[50000x20] = H[50000x256] @ W[256x20], fp32 WMMA, N padded to 32.
// Block = 64 threads (2 waves); wave w covers cols [16w, 16w+16).
// ---------------------------------------------------------------------------
__global__ void __launch_bounds__(64)
gcn_gemm2_wmma(const float* __restrict__ H, const float* __restrict__ W,
               float* __restrict__ C) {
    const int lane    = threadIdx.x & 31;
    const int wave    = threadIdx.x >> 5;                  // 0..1
    const int rowBase = blockIdx.x * 16;
    const int colBase = wave * 16;
    const int m       = lane & 15;
    const int kOff    = (lane >> 4) << 1;
    const int cn      = colBase + m;
    const bool cok    = (cn < NCLS);

    v8f acc = {};
    const float* hrow = H + (size_t)(rowBase + m) * NHID;

    for (int k = 0; k < NHID; k += 4) {
        const int ka = k + kOff;
        v2f a;
        a.x = hrow[ka];
        a.y = hrow[ka + 1];
        v2f b;
        b.x = cok ? W[(size_t)ka * NCLS + cn] : 0.0f;
        b.y = cok ? W[(size_t)(ka + 1) * NCLS + cn] : 0.0f;
        acc = __builtin_amdgcn_wmma_f32_16x16x4_f32(
            false, a, false, b, (short)0, acc, false, false);
    }

    if (cok) {
        const int mOut = (lane >> 4) * 8;
        #pragma unroll
        for (int r = 0; r < 8; ++r) {
            C[(size_t)(rowBase + mOut + r) * NCLS + cn] = acc[r];
        }
    }
}

// ---------------------------------------------------------------------------
// SpMM scatter, 256-wide features: 64 threads per edge, float4 gather,
// 4 scalar f32 atomics per thread. Source and dest rows are L2-resident.
// ---------------------------------------------------------------------------
__global__ void __launch_bounds__(256)
gcn_spmm_f256(const int* __restrict__ rows, const int* __restrict__ cols,
              const float* __restrict__ vals, const float* __restrict__ src,
              float* __restrict__ dst, int n_edges) {
    const int e = blockIdx.x * 4 + (threadIdx.x >> 6);
    if (e >= n_edges) return;
    const int t = threadIdx.x & 63;
    const int r = rows[e];
    const int c = cols[e];
    const float v = vals[e];
    const float4 s = ((const float4*)(src + (size_t)c * NHID))[t];
    float* o = dst + (size_t)r * NHID + t * 4;
    atomicAdd(o + 0, v * s.x);
    atomicAdd(o + 1, v * s.y);
    atomicAdd(o + 2, v * s.z);
    atomicAdd(o + 3, v * s.w);
}

// ---------------------------------------------------------------------------
// SpMM scatter, 20-wide features: 32 threads per edge, lanes 0..19 active.
// ---------------------------------------------------------------------------
__global__ void __launch_bounds__(256)
gcn_spmm_f20(const int* __restrict__ rows, const int* __restrict__ cols,
             const float* __restrict__ vals, const float* __restrict__ src,
             float* __restrict__ dst, int n_edges) {
    const int e = blockIdx.x * 8 + (threadIdx.x >> 5);
    if (e >= n_edges) return;
    const int f = threadIdx.x & 31;
    if (f >= NCLS) return;
    const int r = rows[e];
    const int c = cols[e];
    const float v = vals[e];
    atomicAdd(dst + (size_t)r * NCLS + f, v * src[(size_t)c * NCLS + f]);
}

// ---------------------------------------------------------------------------
// h = relu(h + b1), elementwise over 50000x256 (grid-stride, float4).
// ---------------------------------------------------------------------------
__global__ void __launch_bounds__(256)
gcn_bias_relu(float* __restrict__ h, const float* __restrict__ b1, int total4) {
    int i = blockIdx.x * blockDim.x + threadIdx.x;
    const int stride = gridDim.x * blockDim.x;
    for (; i < total4; i += stride) {
        float4 x = ((float4*)h)[i];
        const int bo = (i * 4) & (NHID - 1);
        x.x = fmaxf(x.x + b1[bo + 0], 0.0f);
        x.y = fmaxf(x.y + b1[bo + 1], 0.0f);
        x.z = fmaxf(x.z + b1[bo + 2], 0.0f);
        x.w = fmaxf(x.w + b1[bo + 3], 0.0f);
        ((float4*)h)[i] = x;
    }
}

// ---------------------------------------------------------------------------
// out[i] = softmax(logits[idx[i]] + b2). One thread per gathered row (tiny).
// ---------------------------------------------------------------------------
__global__ void __launch_bounds__(256)
gcn_softmax_gather(const float* __restrict__ logits, const float* __restrict__ b2,
                   const int* __restrict__ idx, float* __restrict__ out, int n) {
    const int i = blockIdx.x * blockDim.x + threadIdx.x;
    if (i >= n) return;
    const int r = idx[i];
    float v[NCLS];
    float mx = -INFINITY;
    #pragma unroll
    for (int j = 0; j < NCLS; ++j) {
        v[j] = logits[(size_t)r * NCLS + j] + b2[j];
        mx = fmaxf(mx, v[j]);
    }
    float s = 0.0f;
    #pragma unroll
    for (int j = 0; j < NCLS; ++j) {
        v[j] = expf(v[j] - mx);
        s += v[j];
    }
    const float inv = 1.0f / s;
    #pragma unroll
    for (int j = 0; j < NCLS; ++j) {
        out[(size_t)i * NCLS + j] = v[j] * inv;
    }
}

// ---------------------------------------------------------------------------
// Orchestration. Workspace layout (floats):
//   [0 .. 12.8M)      xW1   (then reused: hW2 at 0, logits at 1.0M)
//   [12.8M .. 25.6M)  h
// Total: 102.4 MB.
// ---------------------------------------------------------------------------
extern "C" void kernel_launch(void* const* d_in, const int* in_sizes, int n_in,
                              void* d_out, int out_size, void* d_ws, size_t ws_size,
                              hipStream_t stream) {
    const float* x       = (const float*)d_in[0];
    const int*   fn_rows = (const int*)  d_in[1];
    const int*   fn_cols = (const int*)  d_in[2];
    const float* fn_vals = (const float*)d_in[3];
    const int*   nf_rows = (const int*)  d_in[4];
    const int*   nf_cols = (const int*)  d_in[5];
    const float* nf_vals = (const float*)d_in[6];
    const int*   idx     = (const int*)  d_in[7];
    const float* W1      = (const float*)d_in[8];
    const float* b1      = (const float*)d_in[9];
    const float* W2      = (const float*)d_in[10];
    const float* b2      = (const float*)d_in[11];
    float*       out     = (float*)d_out;

    float* ws     = (float*)d_ws;
    float* xW1    = ws;                                  // 50000*256
    float* h      = ws + (size_t)N_NODES * NHID;         // 50000*256
    float* hW2    = ws;                                  // 50000*20 (reuse xW1)
    float* logits = ws + (size_t)N_NODES * NCLS;         // 50000*20

    // gc1: xW1 = x @ W1   (fp32 WMMA)
    gcn_gemm1_wmma<<<dim3(N_NODES / 16, NHID / 128), 256, 0, stream>>>(x, W1, xW1);

    // h = FN @ xW1 (scatter-add), then + b1, relu
    hipMemsetAsync(h, 0, (size_t)M_NODES * NHID * sizeof(float), stream);
    gcn_spmm_f256<<<N_EDGES / 4, 256, 0, stream>>>(fn_rows, fn_cols, fn_vals,
                                                   xW1, h, N_EDGES);
    gcn_bias_relu<<<2048, 256, 0, stream>>>(h, b1, (M_NODES * NHID) / 4);

    // gc2: hW2 = h @ W2   (fp32 WMMA, N=20 padded to 32)
    gcn_gemm2_wmma<<<M_NODES / 16, 64, 0, stream>>>(h, W2, hW2);

    // logits = NF @ hW2 (scatter-add)
    hipMemsetAsync(logits, 0, (size_t)N_NODES * NCLS * sizeof(float), stream);
    gcn_spmm_f20<<<N_EDGES / 8, 256, 0, stream>>>(nf_rows, nf_cols, nf_vals,
                                                  hW2, logits, N_EDGES);

    // out = softmax(logits + b2)[idx]
    gcn_softmax_gather<<<(N_IDX + 255) / 256, 256, 0, stream>>>(logits, b2, idx,
                                                                out, N_IDX);
}